// CRF_13237089206508
// MI455X (gfx1250) — compile-verified
//
#include <hip/hip_runtime.h>
#include <math.h>

typedef float v2f __attribute__((ext_vector_type(2)));
typedef float v8f __attribute__((ext_vector_type(8)));

#define B_ 64
#define T_ 512
#define D_ 1024
#define L_ 32

// ---------------------------------------------------------------------------
// Kernel 1: emissions = x @ W + b  via V_WMMA_F32_16X16X4_F32
//   grid: (B*T)/64 blocks of 128 threads (4 waves); wave -> 16x32 tile.
//   x tile staged in LDS (stride 68 dwords -> conflict-free b64 A reads).
//   W k-slice staged transposed in LDS (stride 68 -> conflict-free b64 B reads).
//   next x tile prefetched via global_prefetch_b8.
// ---------------------------------------------------------------------------
__global__ __launch_bounds__(128) void emis_gemm(
    const float* __restrict__ X, const float* __restrict__ W,
    const float* __restrict__ bias, float* __restrict__ em)
{
  __shared__ float lx[64 * 68];    // x tile, 64 rows x 64 k (pad 68)
  __shared__ float wtT[32 * 68];   // W^T slice, 32 labels x 64 k (pad 68)
  const int tid  = threadIdx.x;
  const int lane = tid & 31;
  const int wv   = tid >> 5;
  const int row0 = blockIdx.x * 64;
  const int nl   = lane & 15;   // M (A) / N (B,D)
  const int kh   = lane >> 4;   // K half-select

  v8f c0 = {0.f, 0.f, 0.f, 0.f, 0.f, 0.f, 0.f, 0.f};
  v8f c1 = {0.f, 0.f, 0.f, 0.f, 0.f, 0.f, 0.f, 0.f};

  const int rcol = (tid & 15) << 2;  // 0..60 step 4
  const int rrow = tid >> 4;         // 0..7

  for (int kb = 0; kb < D_; kb += 64) {
    __syncthreads();
    // stage 64x64 x tile (coalesced b128)
    #pragma unroll
    for (int p = 0; p < 8; ++p) {
      int r = (p << 3) + rrow;
      float4 v = *(const float4*)&X[(size_t)(row0 + r) * D_ + kb + rcol];
      *(float4*)&lx[r * 68 + rcol] = v;
    }
    // stage transposed W slice: wtT[n][k] = W[kb+k][n] (coalesced b32 reads)
    #pragma unroll
    for (int q = 0; q < 16; ++q) {
      int e = tid + q * 128;   // 0..2047
      int k = e >> 5;          // 0..63
      int n = e & 31;
      wtT[n * 68 + k] = W[(size_t)(kb + k) * L_ + n];
    }
    // prefetch next x tile (one 128B line per thread covers 64 rows x 256B)
    if (kb + 64 < D_) {
      const float* pf =
          &X[(size_t)(row0 + (tid >> 1)) * D_ + kb + 64 + (tid & 1) * 32];
      __builtin_prefetch(pf, 0, 1);
    }
    __syncthreads();
    #pragma unroll
    for (int ks = 0; ks < 64; ks += 4) {
      // A fragment (16x4 f32): lane m=nl holds K = 2*kh + {0,1}
      v2f a  = *(const v2f*)&lx[(wv * 16 + nl) * 68 + ks + 2 * kh];
      // B fragments (4x16 f32): lane n holds K = 2*kh + {0,1} (from W^T in LDS)
      v2f b0 = *(const v2f*)&wtT[nl * 68 + ks + 2 * kh];
      v2f b1 = *(const v2f*)&wtT[(nl + 16) * 68 + ks + 2 * kh];
      c0 = __builtin_amdgcn_wmma_f32_16x16x4_f32(false, a, false, b0,
                                                 (short)0, c0, false, false);
      c1 = __builtin_amdgcn_wmma_f32_16x16x4_f32(false, a, false, b1,
                                                 (short)0, c1, false, false);
    }
  }

  const float bn0 = bias[nl];
  const float bn1 = bias[16 + nl];
  // D layout: VGPR r, lanes 0-15 -> M=r, lanes 16-31 -> M=r+8, N=lane&15
  #pragma unroll
  for (int r = 0; r < 8; ++r) {
    int row = row0 + wv * 16 + r + 8 * kh;
    em[(size_t)row * L_ + nl]      = c0[r] + bn0;
    em[(size_t)row * L_ + 16 + nl] = c1[r] + bn1;
  }
}

// ---------------------------------------------------------------------------
// Kernel 2: fused CRF forward (log-norm), Viterbi (+backtrack), sequence score
//   one wave (32 lanes) per batch; lane = label index
// ---------------------------------------------------------------------------
__global__ __launch_bounds__(32) void crf_scan(
    const float* __restrict__ em, const float* __restrict__ trans,
    const int* __restrict__ label, const int* __restrict__ seqlen,
    float* __restrict__ lognorm, float* __restrict__ score,
    float* __restrict__ vit)
{
  __shared__ unsigned char bp[(T_ - 1) * L_];   // 16 KB backpointers in LDS
  const int b    = blockIdx.x;
  const int lane = threadIdx.x;
  const int sl   = seqlen[b];

  // transitions column trans[j][lane] in 32 VGPRs
  float tc[L_];
  #pragma unroll
  for (int j = 0; j < L_; ++j) tc[j] = trans[j * L_ + lane];

  float alpha = em[((size_t)b * T_ + 0) * L_ + lane];  // log-norm alpha
  float va    = alpha;                                 // viterbi alpha

  for (int t = 1; t < T_; ++t) {
    const float emit = em[((size_t)b * T_ + t) * L_ + lane];
    float m = -3.4e38f, vb = -3.4e38f;
    int vbi = 0;
    float sv[L_];
    #pragma unroll
    for (int j = 0; j < L_; ++j) {
      float aj = __shfl(alpha, j, 32);
      float vj = __shfl(va, j, 32);
      float s = aj + tc[j];
      sv[j] = s;
      m = fmaxf(m, s);
      float v = vj + tc[j];
      if (v > vb) { vb = v; vbi = j; }
    }
    float acc = 0.f;
    #pragma unroll
    for (int j = 0; j < L_; ++j) acc += __expf(sv[j] - m);
    const bool valid = (t < sl);
    float lse  = m + __logf(acc) + emit;
    float vnew = vb + emit;
    alpha = valid ? lse : alpha;
    va    = valid ? vnew : va;
    bp[(t - 1) * L_ + lane] = (unsigned char)(valid ? vbi : lane);
  }

  // log-norm: logsumexp over lanes
  float m2 = alpha;
  #pragma unroll
  for (int off = 16; off >= 1; off >>= 1) m2 = fmaxf(m2, __shfl_xor(m2, off, 32));
  float e = __expf(alpha - m2);
  #pragma unroll
  for (int off = 16; off >= 1; off >>= 1) e += __shfl_xor(e, off, 32);
  const float ln = m2 + __logf(e);

  // viterbi terminal argmax (lowest index on ties, like jnp.argmax)
  float bv = va; int bi = lane;
  #pragma unroll
  for (int off = 16; off >= 1; off >>= 1) {
    float ov = __shfl_xor(bv, off, 32);
    int   oi = __shfl_xor(bi, off, 32);
    if (ov > bv || (ov == bv && oi < bi)) { bv = ov; bi = oi; }
  }

  // gold sequence score (lanes strided over t)
  float sc = 0.f;
  for (int t = lane; t < T_; t += 32) {
    int lt = label[(size_t)b * T_ + t];
    if (t < sl) {
      sc += em[((size_t)b * T_ + t) * L_ + lt];
      if (t >= 1) sc += trans[label[(size_t)b * T_ + t - 1] * L_ + lt];
    }
  }
  #pragma unroll
  for (int off = 16; off >= 1; off >>= 1) sc += __shfl_xor(sc, off, 32);

  __syncthreads();  // LDS bp visibility for lane 0

  if (lane == 0) {
    lognorm[b] = ln;
    score[b]   = sc;
    int tag = bi;
    for (int t = T_ - 1; t >= 1; --t) {
      vit[(size_t)b * T_ + t] = (float)tag;
      tag = (int)bp[(t - 1) * L_ + tag];
    }
    vit[(size_t)b * T_] = (float)tag;
  }
}

// ---------------------------------------------------------------------------
// Kernel 3: loss + tp/tn/fp (deterministic single-block tree reduction)
// ---------------------------------------------------------------------------
__global__ __launch_bounds__(256) void finalize(
    const float* __restrict__ lognorm, const float* __restrict__ score,
    const float* __restrict__ vit, const int* __restrict__ label,
    const int* __restrict__ seqlen, float* __restrict__ out)
{
  __shared__ float r0[256], r1[256], r2[256], r3[256];
  const int tid = threadIdx.x;

  float lp = 0.f;
  for (int b = tid; b < B_; b += 256) lp += lognorm[b] - score[b];

  float tp = 0.f, tn = 0.f, fp = 0.f;
  for (int i = tid; i < B_ * T_; i += 256) {
    int b = i / T_, t = i % T_;
    int l = label[i];
    int v = (int)vit[i];
    if (l > 0) { if (v == l) tp += 1.f; else tn += 1.f; }
    if (t < seqlen[b] && l == 0 && v > 0) fp += 1.f;
  }

  r0[tid] = lp; r1[tid] = tp; r2[tid] = tn; r3[tid] = fp;
  __syncthreads();
  for (int s = 128; s >= 1; s >>= 1) {
    if (tid < s) {
      r0[tid] += r0[tid + s];
      r1[tid] += r1[tid + s];
      r2[tid] += r2[tid + s];
      r3[tid] += r3[tid + s];
    }
    __syncthreads();
  }
  if (tid == 0) {
    out[0] = r0[0] / (float)B_;                 // loss = mean(-ll)
    out[1 + B_ * T_ + 0] = r1[0];               // tp
    out[1 + B_ * T_ + 1] = r2[0];               // tn
    out[1 + B_ * T_ + 2] = r3[0];               // fp
  }
}

// ---------------------------------------------------------------------------
extern "C" void kernel_launch(void* const* d_in, const int* in_sizes, int n_in,
                              void* d_out, int out_size, void* d_ws, size_t ws_size,
                              hipStream_t stream) {
  const float* x     = (const float*)d_in[0];
  const float* W     = (const float*)d_in[1];
  const float* bias  = (const float*)d_in[2];
  const float* trans = (const float*)d_in[3];
  const int*   label = (const int*)d_in[4];
  const int*   slen  = (const int*)d_in[5];
  // d_in[6] (mask) recomputed from seqlen; not read.

  float* out = (float*)d_out;            // [loss, vit(B*T), tp, tn, fp]
  float* em      = (float*)d_ws;         // B*T*L floats = 4 MB
  float* lognorm = em + (size_t)B_ * T_ * L_;
  float* score   = lognorm + B_;

  emis_gemm<<<(B_ * T_) / 64, 128, 0, stream>>>(x, W, bias, em);
  crf_scan<<<B_, 32, 0, stream>>>(em, trans, label, slen, lognorm, score, out + 1);
  finalize<<<1, 256, 0, stream>>>(lognorm, score, out + 1, label, slen, out);
}